// EntropyPatcher_38156489457732
// MI455X (gfx1250) — compile-verified
//
#include <hip/hip_runtime.h>
#include <stdint.h>

#define B 128
#define S 8192
#define LOOKAHEAD 8
#define THRESHOLD 2.9f
#define MAXP 4096          // ceil(S / MIN_PATCH)
#define MAX_PATCH 16
#define SEG 128
#define NSEG (S / SEG)     // 64
#define TPB 256            // 8 wave32 waves

typedef unsigned int u32x4 __attribute__((ext_vector_type(4)));
typedef int          i32x8 __attribute__((ext_vector_type(8)));
typedef int          i32x4 __attribute__((ext_vector_type(4)));
typedef float        f32x4 __attribute__((ext_vector_type(4)));

// ---------------------------------------------------------------------------
// Kernel A: one block per batch row. TDM-stages the token row into LDS,
// computes entropy -> stop -> next_stop (suffix min) -> sizes, then runs the
// segmented two-pass chain walk and writes packed (start|size<<13) records.
// ---------------------------------------------------------------------------
__global__ __launch_bounds__(TPB)
void patch_build_kernel(const int* __restrict__ tokens,
                        unsigned int* __restrict__ recs) {
  __shared__ int            s_tok[S];            // 32 KB (TDM destination)
  __shared__ unsigned short s_nstop[S + 8];      // next_stop, 16 KB
  __shared__ unsigned char  s_sz[S];             // patch size per start, 8 KB
  __shared__ unsigned short s_steps[NSEG * 16];  // pass-A step counts
  __shared__ unsigned char  s_exit[NSEG * 16];   // pass-A exit offsets
  __shared__ unsigned short s_cmin[TPB];
  __shared__ unsigned short s_suf[TPB];
  __shared__ unsigned char  s_entC[NSEG];
  __shared__ unsigned short s_rnkC[NSEG];

  const int tid = threadIdx.x;
  const int b   = blockIdx.x;
  const int* row = tokens + (size_t)b * S;
  unsigned int* rrow = recs + (size_t)b * MAXP;

  // Phase 0: sentinel-clear this row's record slots (overlaps the DMA below).
  for (int p = tid; p < MAXP; p += TPB) rrow[p] = 0xFFFFFFFFu;

  // Stage the 32KB token row Global -> LDS via the Tensor Data Mover.
#if __has_builtin(__builtin_amdgcn_tensor_load_to_lds) && __has_builtin(__builtin_amdgcn_s_wait_tensorcnt)
  if (tid < 32) {  // one wave issues the TDM descriptor (EXEC ignored by TDM)
    uint64_t ga  = (uint64_t)(uintptr_t)row;
    uint32_t lds = (uint32_t)(uintptr_t)&s_tok[0];  // low 32 bits = LDS byte offset
    u32x4 g0;
    g0[0] = 1u;                                        // count=1, user descriptor
    g0[1] = lds;                                       // lds_addr
    g0[2] = (uint32_t)(ga & 0xFFFFFFFFu);              // global_addr[31:0]
    g0[3] = (uint32_t)((ga >> 32) & 0x01FFFFFFu)       // global_addr[56:32]
          | (2u << 30);                                // type = 2 ("image")
    i32x8 g1;
    g1[0] = (int)(2u << 16);                           // data_size = 4 bytes
    g1[1] = (int)(((unsigned)S & 0xFFFFu) << 16);      // tensor_dim0 lo -> bits[63:48]
    g1[2] = (int)(((unsigned)S >> 16) | (1u << 16));   // tensor_dim0 hi, tensor_dim1 = 1
    g1[3] = (int)((unsigned)S << 16);                  // tile_dim0 = S -> bits[127:112]
    g1[4] = 1;                                         // tile_dim1 = 1
    g1[5] = S;                                         // tensor_dim0_stride = S
    g1[6] = 0;
    g1[7] = 0;
    i32x4 gz4 = {0, 0, 0, 0};                          // groups 2/3 unused (<=2D)
    i32x8 gz8 = {0, 0, 0, 0, 0, 0, 0, 0};
    __builtin_amdgcn_tensor_load_to_lds(g0, g1, gz4, gz4, gz8, 0);
    __builtin_amdgcn_s_wait_tensorcnt(0);
  }
#else
  for (int i = tid; i < S; i += TPB) s_tok[i] = row[i];
#endif
  __syncthreads();

  // Phase 1: windowed entropy; stop[j] = (ent >= THRESHOLD); seed suffix-min.
  for (int j = tid; j < S; j += TPB) {
    int L = S - j; if (L > LOOKAHEAD) L = LOOKAHEAD;
    int w[LOOKAHEAD];
#pragma unroll
    for (int k = 0; k < LOOKAHEAD; ++k) {
      int idx = j + k; if (idx > S - 1) idx = S - 1;
      w[k] = s_tok[idx];
    }
    float invL = 1.0f / (float)L;
    float acc = 0.0f;
    for (int k = 0; k < L; ++k) {
      int c = 0;
      for (int m = 0; m < L; ++m) c += (w[k] == w[m]) ? 1 : 0;
      acc += __log2f((float)c * invL);
    }
    float ent = -acc * invL;
    s_nstop[j] = (ent >= THRESHOLD) ? (unsigned short)j : (unsigned short)S;
  }
  if (tid == 0) s_nstop[S] = (unsigned short)S;
  __syncthreads();

  // Phase 2: reverse suffix-min over S entries (chunked 32 per thread).
  {
    const int CH = S / TPB;  // 32
    int base = tid * CH;
    int m = S;
    for (int k = CH - 1; k >= 0; --k) {
      int v = s_nstop[base + k];
      if (v < m) m = v;
      s_nstop[base + k] = (unsigned short)m;
    }
    s_cmin[tid] = (unsigned short)m;
  }
  __syncthreads();
  if (tid == 0) {
    int m = S;
    for (int c = TPB - 1; c >= 0; --c) {
      s_suf[c] = (unsigned short)m;
      int v = s_cmin[c];
      if (v < m) m = v;
    }
  }
  __syncthreads();
  {
    const int CH = S / TPB;
    int base = tid * CH;
    int a = s_suf[tid];
    for (int k = 0; k < CH; ++k) {
      if (a < (int)s_nstop[base + k]) s_nstop[base + k] = (unsigned short)a;
    }
  }
  __syncthreads();

  // Phase 3: sizes[i] = min(MAX_PATCH, next_stop[min(i+2,S)] - i)  (always >=1)
  for (int i = tid; i < S; i += TPB) {
    int ip = i + 2; if (ip > S) ip = S;
    int szv = (int)s_nstop[ip] - i;
    if (szv > MAX_PATCH) szv = MAX_PATCH;
    s_sz[i] = (unsigned char)szv;
  }
  __syncthreads();

  // Pass A: for every (segment, entry offset in [0,16)) walk to segment exit.
  for (int wid = tid; wid < NSEG * 16; wid += TPB) {
    int sg = wid >> 4, e = wid & 15;
    int q = sg * SEG + e;
    int end = (sg + 1) * SEG;
    int steps = 0;
    while (q < end) { steps++; q += (int)s_sz[q]; }
    s_steps[wid] = (unsigned short)steps;
    s_exit[wid]  = (unsigned char)(q - end);
  }
  __syncthreads();

  // Compose segment entries/ranks (64 serial steps; jumps <=16 so every
  // segment is entered at an offset < 16).
  if (tid == 0) {
    int entry = 0, rank = 0;
    for (int sg = 0; sg < NSEG; ++sg) {
      s_entC[sg] = (unsigned char)entry;
      s_rnkC[sg] = (unsigned short)rank;
      int wid = (sg << 4) | entry;
      rank += (int)s_steps[wid];
      entry = (int)s_exit[wid];
    }
  }
  __syncthreads();

  // Pass B: replay all segments in parallel at known ranks; emit records.
  if (tid < NSEG) {
    int q    = tid * SEG + (int)s_entC[tid];
    int rank = (int)s_rnkC[tid];
    int end  = (tid + 1) * SEG;
    while (q < end) {
      unsigned int szv = (unsigned int)s_sz[q];
      rrow[rank] = (unsigned int)q | (szv << 13);   // start:13b | size:5b
      rank++;
      q += (int)szv;
    }
  }
}

// ---------------------------------------------------------------------------
// Kernel B: one thread per patch. Gathers <=16 tokens (L2-resident) and
// streams the 36MB output with non-temporal b128 stores.
// ---------------------------------------------------------------------------
__global__ __launch_bounds__(TPB)
void patch_write_kernel(const int* __restrict__ tokens,
                        const unsigned int* __restrict__ recs,
                        float* __restrict__ out) {
  const long long gid = (long long)blockIdx.x * TPB + threadIdx.x;
  if (gid >= (long long)B * MAXP) return;
  const int b = (int)(gid >> 12);  // MAXP == 4096

  unsigned int rec = recs[gid];
  float vals[MAX_PATCH];
  float mval;
  if (rec != 0xFFFFFFFFu) {
    int start = (int)(rec & 0x1FFFu);
    int sz    = (int)((rec >> 13) & 0x1Fu);
    const int* row = tokens + (size_t)b * S;
#pragma unroll
    for (int l = 0; l < MAX_PATCH; ++l) {
      int idx = start + l; if (idx > S - 1) idx = S - 1;
      vals[l] = (l < sz) ? (float)row[idx] : 0.0f;
    }
    mval = 1.0f;
  } else {
#pragma unroll
    for (int l = 0; l < MAX_PATCH; ++l) vals[l] = 0.0f;
    mval = 0.0f;
  }

  f32x4* dst = (f32x4*)(out + (size_t)gid * MAX_PATCH);
#pragma unroll
  for (int v = 0; v < 4; ++v) {
    f32x4 f4 = { vals[4 * v + 0], vals[4 * v + 1], vals[4 * v + 2], vals[4 * v + 3] };
    __builtin_nontemporal_store(f4, dst + v);
  }
  float* mout = out + (size_t)B * MAXP * MAX_PATCH;
  __builtin_nontemporal_store(mval, mout + gid);
}

extern "C" void kernel_launch(void* const* d_in, const int* in_sizes, int n_in,
                              void* d_out, int out_size, void* d_ws, size_t ws_size,
                              hipStream_t stream) {
  const int* tokens  = (const int*)d_in[0];
  unsigned int* recs = (unsigned int*)d_ws;        // B*MAXP*4 = 2 MB scratch
  float* out         = (float*)d_out;              // patches then mask, flat

  patch_build_kernel<<<B, TPB, 0, stream>>>(tokens, recs);

  const int total = B * MAXP;
  patch_write_kernel<<<(total + TPB - 1) / TPB, TPB, 0, stream>>>(tokens, recs, out);
}